// DeformConv2d_74302934221063
// MI455X (gfx1250) — compile-verified
//
#include <hip/hip_runtime.h>
#include <hip/hip_bf16.h>
#include <math.h>

typedef __attribute__((ext_vector_type(2))) float v2f;
typedef __attribute__((ext_vector_type(8))) float v8f;

#define B_    4
#define C_    256
#define OC_   256
#define H_    64
#define W_    64
#define HW_   (H_ * W_)          // 4096
#define NPOS_ (B_ * HW_)         // 16384
#define N_    9
#define K_    (N_ * C_)          // 2304
#define SROW  260                // padded LDS row stride (260 % 64 == 4)

__device__ __forceinline__ float sigmoidf_(float v) { return 1.0f / (1.0f + __expf(-v)); }

// ---------------------------------------------------------------------------
// Kernel 1: the three small 3x3 convs (w_p:18ch, w_m:9ch, w_ad:3ch), pad=1.
// conv_out layout: [och(30)][pos(16384)]
// ---------------------------------------------------------------------------
__global__ void __launch_bounds__(256) small_convs_kernel(
    const float* __restrict__ x,
    const float* __restrict__ w_p, const float* __restrict__ b_p,
    const float* __restrict__ w_m, const float* __restrict__ b_m,
    const float* __restrict__ w_ad, const float* __restrict__ b_ad,
    float* __restrict__ conv_out)
{
    int t   = blockIdx.x * 256 + threadIdx.x;   // 30 * 16384 threads
    int och = t >> 14;
    int pos = t & (NPOS_ - 1);
    int bb  = pos >> 12;
    int sp  = pos & (HW_ - 1);
    int i   = sp >> 6;
    int j   = sp & 63;

    const float* w;
    float acc;
    if (och < 18)      { w = w_p  + och * (C_ * 9);        acc = b_p[och];       }
    else if (och < 27) { w = w_m  + (och - 18) * (C_ * 9); acc = b_m[och - 18];  }
    else               { w = w_ad + (och - 27) * (C_ * 9); acc = b_ad[och - 27]; }

    const float* xb = x + (size_t)bb * C_ * HW_;
    for (int c = 0; c < C_; ++c) {
        const float* xc = xb + (size_t)c * HW_;
        const float* wc = w + c * 9;
        #pragma unroll
        for (int ki = 0; ki < 3; ++ki) {
            int ii = i + ki - 1;
            if (ii < 0 || ii >= H_) continue;
            #pragma unroll
            for (int kj = 0; kj < 3; ++kj) {
                int jj = j + kj - 1;
                if (jj < 0 || jj >= W_) continue;
                acc = fmaf(xc[ii * W_ + jj], wc[ki * 3 + kj], acc);
            }
        }
    }
    conv_out[(size_t)och * NPOS_ + pos] = acc;
}

// ---------------------------------------------------------------------------
// Kernel 2: repack w_conv (OC, C, 3, 3) -> wT2[kpair = k/2][oc][k&1] with
// k = n*256 + c. A lane's A-fragment (A[m][k], A[m][k+1]) becomes one
// aligned 8-byte load; lanes 0-15 cover a contiguous 128 B segment.
// ---------------------------------------------------------------------------
__global__ void __launch_bounds__(256) wt_kernel(
    const float* __restrict__ w_conv, float* __restrict__ wT2)
{
    int t = blockIdx.x * 256 + threadIdx.x;   // K_ * OC_ threads
    int k = t >> 8;
    int m = t & 255;
    if (k >= K_) return;
    int n = k >> 8;     // k = n*256 + c
    int c = k & 255;
    wT2[(size_t)(k >> 1) * (2 * OC_) + m * 2 + (k & 1)] =
        w_conv[(size_t)m * K_ + c * 9 + n];
}

// ---------------------------------------------------------------------------
// Kernel 3: per (pos, n) bilinear sampling metadata, with mask*ad_m folded
// into the 4 tap weights. meta entry = {int idx[4]; float w[4];} = 32 bytes.
// ---------------------------------------------------------------------------
__global__ void __launch_bounds__(256) meta_kernel(
    const float* __restrict__ conv_out, char* __restrict__ meta)
{
    int t = blockIdx.x * 256 + threadIdx.x;   // NPOS_ * 9 threads
    int pos = t / 9;
    int n   = t - pos * 9;
    int sp  = pos & (HW_ - 1);
    int i   = sp >> 6;
    int j   = sp & 63;

    float offx = conv_out[(size_t)n * NPOS_ + pos];
    float offy = conv_out[(size_t)(9 + n) * NPOS_ + pos];
    float mval = sigmoidf_(conv_out[(size_t)(18 + n) * NPOS_ + pos]);
    float adb  = 1.0f - sigmoidf_(conv_out[(size_t)(27 + (n % 3)) * NPOS_ + pos]);
    float ad   = adb * 2.0f;                  // tile(ad_base) * DIL
    float adm  = (adb - 0.5f) * 2.0f * 2.0f;  // (ad_base-0.5)*2 * DIL

    float rx = (float)(n / 3) - 1.0f;         // repeat(r,3)
    float ry = (float)(n % 3) - 1.0f;         // tile(r,3)

    float px = (1.0f + (float)i) + rx + offx + ad * rx;
    float py = (1.0f + (float)j) + ry + offy + ad * ry;

    const float HPm1 = 65.0f;                 // Hp-1 = 66-1
    float fx = floorf(px), fy = floorf(py);
    float qltx = fminf(fmaxf(fx, 0.0f), HPm1);
    float qlty = fminf(fmaxf(fy, 0.0f), HPm1);
    float qrbx = fminf(fmaxf(fx + 1.0f, 0.0f), HPm1);
    float qrby = fminf(fmaxf(fy + 1.0f, 0.0f), HPm1);

    bool mx = (px < 1.0f) || (px > 64.0f);
    bool my = (py < 1.0f) || (py > 64.0f);
    float pxc = fminf(fmaxf(mx ? fx : px, 0.0f), HPm1);
    float pyc = fminf(fmaxf(my ? fy : py, 0.0f), HPm1);

    float glt = (1.0f + qltx - pxc) * (1.0f + qlty - pyc);
    float grb = (1.0f - qrbx + pxc) * (1.0f - qrby + pyc);
    float glb = (1.0f + qltx - pxc) * (1.0f - qrby + pyc);
    float grt = (1.0f - qrbx + pxc) * (1.0f + qlty - pyc);

    float mm = mval * adm;

    float tqx[4] = { qltx, qrbx, qltx, qrbx };
    float tqy[4] = { qlty, qrby, qrby, qlty };
    float tg [4] = { glt,  grb,  glb,  grt  };

    int   idx[4];
    float wgt[4];
    #pragma unroll
    for (int tt = 0; tt < 4; ++tt) {
        int ux = (int)tqx[tt] - 1;            // padded -> unpadded coords
        int uy = (int)tqy[tt] - 1;
        if (ux < 0 || ux >= H_ || uy < 0 || uy >= W_) {
            idx[tt] = 0; wgt[tt] = 0.0f;      // padded x is zero there
        } else {
            idx[tt] = ux * W_ + uy;
            wgt[tt] = tg[tt] * mm;
        }
    }

    int4*  mo = (int4*)(meta + (size_t)t * 32);
    int4   qi = make_int4(idx[0], idx[1], idx[2], idx[3]);
    float4 qw = make_float4(wgt[0], wgt[1], wgt[2], wgt[3]);
    mo[0] = qi;
    ((float4*)mo)[1] = qw;
}

// ---------------------------------------------------------------------------
// Kernel 4: fused gather + GEMM via V_WMMA_F32_16X16X4_F32.
// One workgroup (8 waves) per 16-position tile; the bilinear-gathered
// 256(c) x 16(pos) x_off panel for each kernel tap n is staged ONCE into a
// double-buffered LDS panel (row-major [pos][c], 260-float padded stride:
// ds_store_b128 staging, conflict-free ds_load_b64 B reads) and consumed by
// all 8 waves (2 oc-tiles each = all 256 output channels). x (16 MB) stays
// fully resident in MI455X's 192 MB L2, so the gathers are L2-served.
// ---------------------------------------------------------------------------
__global__ void __launch_bounds__(256) dconv_gemm_kernel(
    const float* __restrict__ x, const float* __restrict__ wT2,
    const char* __restrict__ meta, float* __restrict__ out)
{
    __shared__ float sbuf[2][16 * SROW];               // 2 x 16.25 KB panels
    __shared__ __align__(16) char lmeta[16 * N_ * 32]; // 4.5 KB tile metadata

    const int t     = threadIdx.x;
    const int lane  = t & 31;
    const int wave  = t >> 5;
    const int ptile = blockIdx.x;        // 0..1023, 16 positions each

    // --- stage this tile's metadata (contiguous 144 * 32 B) into LDS ---
    if (t < 16 * N_) {
        const int4* s = (const int4*)(meta + ((size_t)ptile * 16 * N_ + t) * 32);
        int4* d = (int4*)(lmeta + t * 32);
        d[0] = s[0];
        d[1] = s[1];
    }
    __syncthreads();

    const int bb = (ptile * 16) >> 12;   // batch is uniform across the tile
    const float* xb = x + (size_t)bb * C_ * HW_;

    // staging mapping: thread -> (pos = t&15, channel group = t>>4)
    const int spos  = t & 15;
    const int cbase = (t >> 4) << 4;     // 16 channels per thread

    // compute mapping (per wave): 2 oc-tiles, ISA 16x4 A / 4x16 B layouts
    const int col  = lane & 15;
    const int half = lane >> 4;          // 0: K0/K1, 1: K2/K3
    const int koff = half << 1;
    const int m0   = wave * 32 + col;    // oc-tile 2*wave
    const int m1   = m0 + 16;            // oc-tile 2*wave+1

    v8f acc0 = {};
    v8f acc1 = {};

    // --- stage tap n = 0 ---
    {
        const int4* mp = (const int4*)(lmeta + (spos * N_ + 0) * 32);
        int4   qi = mp[0];
        float4 qw = ((const float4*)mp)[1];
        float* dst = &sbuf[0][spos * SROW + cbase];
        #pragma unroll
        for (int g = 0; g < 4; ++g) {
            float tmp[4];
            #pragma unroll
            for (int e = 0; e < 4; ++e) {
                const float* xc = xb + (size_t)(cbase + g * 4 + e) * HW_;
                tmp[e] = qw.x * xc[qi.x] + qw.y * xc[qi.y] +
                         qw.z * xc[qi.z] + qw.w * xc[qi.w];
            }
            *(float4*)(dst + g * 4) = make_float4(tmp[0], tmp[1], tmp[2], tmp[3]);
        }
    }

    for (int n = 0; n < N_; ++n) {
        __syncthreads();                 // panel n ready; panel n-1 consumed

        // --- stage tap n+1 into the other buffer (overlaps with compute) ---
        if (n + 1 < N_) {
            const int4* mp = (const int4*)(lmeta + (spos * N_ + (n + 1)) * 32);
            int4   qi = mp[0];
            float4 qw = ((const float4*)mp)[1];
            float* dst = &sbuf[(n + 1) & 1][spos * SROW + cbase];
            #pragma unroll
            for (int g = 0; g < 4; ++g) {
                float tmp[4];
                #pragma unroll
                for (int e = 0; e < 4; ++e) {
                    const float* xc = xb + (size_t)(cbase + g * 4 + e) * HW_;
                    tmp[e] = qw.x * xc[qi.x] + qw.y * xc[qi.y] +
                             qw.z * xc[qi.z] + qw.w * xc[qi.w];
                }
                *(float4*)(dst + g * 4) = make_float4(tmp[0], tmp[1], tmp[2], tmp[3]);
            }
        }

        // --- WMMA sweep over c for this tap ---
        const float* row = &sbuf[n & 1][col * SROW];
        const int kn = n << 8;
        #pragma unroll 4
        for (int c = 0; c < C_; c += 4) {
            const int k = kn + c + koff;                  // even
            const float* wp = wT2 + (size_t)(k >> 1) * (2 * OC_);
            // A pairs: one b64 per oc-tile
            v2f a0 = *(const v2f*)(wp + m0 * 2);
            v2f a1 = *(const v2f*)(wp + m1 * 2);
            // B 4x16: lanes 0-15 rows K+0/K+1, lanes 16-31 rows K+2/K+3
            v2f bv = *(const v2f*)(row + c + koff);
            acc0 = __builtin_amdgcn_wmma_f32_16x16x4_f32(
                false, a0, false, bv, (short)0, acc0, false, false);
            acc1 = __builtin_amdgcn_wmma_f32_16x16x4_f32(
                false, a1, false, bv, (short)0, acc1, false, false);
        }
    }

    // D 16x16 f32 layout: VGPR v -> M = v + half*8, N = lane&15
    const int sp = (ptile * 16 + col) & (HW_ - 1);
    float* ob = out + (size_t)bb * OC_ * HW_ + sp;
    #pragma unroll
    for (int v = 0; v < 8; ++v) {
        int oc0 = wave * 32 + v + (half << 3);
        ob[(size_t)oc0 * HW_]        = acc0[v];
        ob[(size_t)(oc0 + 16) * HW_] = acc1[v];
    }
}

// ---------------------------------------------------------------------------
extern "C" void kernel_launch(void* const* d_in, const int* in_sizes, int n_in,
                              void* d_out, int out_size, void* d_ws, size_t ws_size,
                              hipStream_t stream)
{
    const float* x      = (const float*)d_in[0];
    const float* w_conv = (const float*)d_in[1];
    const float* w_p    = (const float*)d_in[2];
    const float* b_p    = (const float*)d_in[3];
    const float* w_m    = (const float*)d_in[4];
    const float* b_m    = (const float*)d_in[5];
    const float* w_ad   = (const float*)d_in[6];
    const float* b_ad   = (const float*)d_in[7];
    float* out = (float*)d_out;

    char*  ws       = (char*)d_ws;
    float* conv_out = (float*)ws;                                   // 30*16384 f32
    char*  meta     = ws + (size_t)30 * NPOS_ * sizeof(float);      // 16384*9*32 B
    float* wT2      = (float*)(meta + (size_t)NPOS_ * 9 * 32);      // 2304*256 f32

    // 1) aux convs: 30 * 16384 threads
    small_convs_kernel<<<(30 * NPOS_) / 256, 256, 0, stream>>>(
        x, w_p, b_p, w_m, b_m, w_ad, b_ad, conv_out);

    // 2) weight repack: K_ * OC_ threads
    wt_kernel<<<(K_ * OC_) / 256, 256, 0, stream>>>(w_conv, wT2);

    // 3) sampling metadata: NPOS_ * 9 threads
    meta_kernel<<<(NPOS_ * 9) / 256, 256, 0, stream>>>(conv_out, meta);

    // 4) fused LDS-staged gather + WMMA GEMM: 1024 blocks, one per 16-pos tile
    dconv_gemm_kernel<<<1024, 256, 0, stream>>>(x, wT2, meta, out);
}